// VQVAE_31439160606921
// MI455X (gfx1250) — compile-verified
//
#include <hip/hip_runtime.h>

typedef __attribute__((ext_vector_type(2))) float v2f;
typedef __attribute__((ext_vector_type(8))) float v8f;

// ---- workspace layout (in floats) ----
#define WS_BUF1   0u            // 16,777,216 : y1 (conv1 out) then y3 (dec1 out)
#define WS_Y2     16777216u     //  1,048,576 : conv2 out
#define WS_H3     17825792u     //  1,048,576 : post-conv out (dec1 input)
#define WS_STATS  18874368u     //      4,096 : per-channel {sum,sumsq} partials
#define WS_SC1    (WS_STATS + 4096u)   // 32 : bn1 scale/shift pairs
#define WS_SC2    (WS_STATS + 4160u)   //  8 : bn2
#define WS_SC3    (WS_STATS + 4224u)   // 32 : bn3
#define WS_LOSSP  (WS_STATS + 4352u)   // 1024 : loss partials
// total need: (18874368 + 5376) * 4 B  ~= 75.5 MB of d_ws

__device__ __forceinline__ int clampi(int v, int lo, int hi) {
    v = v < lo ? lo : v;
    return v > hi ? hi : v;
}

// ================= conv1: 1->16, k4 s2 p1, in 256x256 -> out 128x128 =================
// GEMM per wave: M=16 (ox tile), N=16 (oc), K=16 (4x4 taps) -> 4x wmma f32 16x16x4
__global__ __launch_bounds__(256) void k_conv1(const float* __restrict__ x,
                                               const float* __restrict__ w1,
                                               const float* __restrict__ b1,
                                               float* __restrict__ ws) {
    __shared__ float lw[256];                 // w1[oc][k], k = ky*4+kx
    int tid = threadIdx.x;
    lw[tid] = w1[tid];
    __syncthreads();
    int wave = blockIdx.x * 8 + (tid >> 5);   // 65536 waves
    int lane = tid & 31;
    int hi = lane >> 4;                       // lane half
    int l  = lane & 15;
    int xt = wave & 7;
    int oy = (wave >> 3) & 127;
    int n  = wave >> 10;
    int ox0 = xt << 4;
    const float* xp = x + n * (256 * 256);
    v8f acc = {};
#pragma unroll
    for (int s = 0; s < 4; ++s) {
        int kb = 4 * s + 2 * hi;
        v2f a, bf;
#pragma unroll
        for (int u = 0; u < 2; ++u) {
            int k  = kb + u;
            int ky = k >> 2, kx = k & 3;
            int iy = 2 * oy - 1 + ky;
            int ix = 2 * (ox0 + l) - 1 + kx;
            bool inb = ((unsigned)iy < 256u) & ((unsigned)ix < 256u);
            float v = xp[clampi(iy, 0, 255) * 256 + clampi(ix, 0, 255)]; // always-valid load
            a[u]  = inb ? v : 0.0f;                                      // branchless pad
            bf[u] = lw[l * 16 + k];
        }
        acc = __builtin_amdgcn_wmma_f32_16x16x4_f32(false, a, false, bf, (short)0, acc, false, false);
    }
    float bias = b1[l];
    float* y1 = ws + WS_BUF1;
    int base = ((n * 16 + l) * 128 + oy) * 128 + ox0;
#pragma unroll
    for (int r = 0; r < 8; ++r) y1[base + r + 8 * hi] = acc[r] + bias;
}

// ================= per-channel stats: deterministic two-level reduction =================
__global__ __launch_bounds__(256) void k_stats(const float* __restrict__ src,
                                               float* __restrict__ partials,
                                               int C, int logPS, int P) {
    __shared__ float ssum[256], ssq[256];
    int c = blockIdx.x / P;
    int p = blockIdx.x - c * P;
    int PS = 1 << logPS;
    int E  = 64 << logPS;                    // 64 batch images per channel
    float s = 0.f, q = 0.f;
    for (int e = p * 256 + threadIdx.x; e < E; e += P * 256) {
        int nn = e >> logPS;
        int ii = e & (PS - 1);
        float v = src[(nn * C + c) * PS + ii];
        s += v; q += v * v;
    }
    ssum[threadIdx.x] = s; ssq[threadIdx.x] = q;
    __syncthreads();
    for (int off = 128; off > 0; off >>= 1) {
        if (threadIdx.x < off) {
            ssum[threadIdx.x] += ssum[threadIdx.x + off];
            ssq[threadIdx.x]  += ssq[threadIdx.x + off];
        }
        __syncthreads();
    }
    if (threadIdx.x == 0) {
        partials[(c * P + p) * 2 + 0] = ssum[0];
        partials[(c * P + p) * 2 + 1] = ssq[0];
    }
}

__global__ void k_bnfin(const float* __restrict__ partials, const float* __restrict__ g,
                        const float* __restrict__ b, float* __restrict__ scsh,
                        int C, int P, float invE) {
    int c = threadIdx.x;
    if (c < C) {
        float s = 0.f, q = 0.f;
        for (int p = 0; p < P; ++p) { s += partials[(c * P + p) * 2]; q += partials[(c * P + p) * 2 + 1]; }
        float m  = s * invE;
        float v  = fmaxf(q * invE - m * m, 0.0f);
        float sc = g[c] * rsqrtf(v + 1e-5f);
        scsh[2 * c]     = sc;
        scsh[2 * c + 1] = b[c] - m * sc;
    }
}

// ================= conv2: 16->4, k4 s2 p1, BN1+ReLU fused on load ====================
// GEMM per wave: M=16, N=16 (cols>=4 zero-padded in LDS), K=256 -> 64x wmma
__global__ __launch_bounds__(256) void k_conv2(const float* __restrict__ w2,
                                               const float* __restrict__ b2,
                                               float* __restrict__ ws) {
    __shared__ float lw[4096];               // lw[oc][k]: oc 0..15, cols 4..15 zero
    __shared__ float lsc[16], lsh[16];
    int tid = threadIdx.x;
    for (int i = tid; i < 4096; i += 256) lw[i] = (i < 1024) ? w2[i] : 0.0f;
    if (tid < 16) { lsc[tid] = ws[WS_SC1 + 2 * tid]; lsh[tid] = ws[WS_SC1 + 2 * tid + 1]; }
    __syncthreads();
    int wave = blockIdx.x * 8 + (tid >> 5);  // 16384 waves
    int lane = tid & 31;
    int hi = lane >> 4;
    int l  = lane & 15;
    int xt = wave & 3;
    int oy = (wave >> 2) & 63;
    int n  = wave >> 8;
    int ox0 = xt << 4;
    const float* y1 = ws + WS_BUF1 + n * (16 * 128 * 128);
    v8f acc = {};
#pragma unroll 4
    for (int s = 0; s < 64; ++s) {
        int kb = 4 * s + 2 * hi;
        v2f a, bf;
#pragma unroll
        for (int u = 0; u < 2; ++u) {
            int k  = kb + u;
            int ic = k >> 4, ky = (k >> 2) & 3, kx = k & 3;
            int iy = 2 * oy - 1 + ky;
            int ix = 2 * (ox0 + l) - 1 + kx;
            bool inb = ((unsigned)iy < 128u) & ((unsigned)ix < 128u);
            float raw = y1[(ic * 128 + clampi(iy, 0, 127)) * 128 + clampi(ix, 0, 127)];
            float v   = fmaxf(raw * lsc[ic] + lsh[ic], 0.0f);   // BN1 + ReLU fused
            a[u]  = inb ? v : 0.0f;
            bf[u] = lw[l * 256 + k];
        }
        acc = __builtin_amdgcn_wmma_f32_16x16x4_f32(false, a, false, bf, (short)0, acc, false, false);
    }
    if (l < 4) {
        float bias = b2[l];
        float* y2 = ws + WS_Y2;
        int base = ((n * 4 + l) * 64 + oy) * 64 + ox0;
#pragma unroll
        for (int r = 0; r < 8; ++r) y2[base + r + 8 * hi] = acc[r] + bias;
    }
}

// ================= fused BN2+ReLU -> pre 1x1 -> VQ -> post 1x1 =======================
__global__ __launch_bounds__(256) void k_vq(const float* __restrict__ pre_w,
                                            const float* __restrict__ pre_b,
                                            const float* __restrict__ cb,
                                            const float* __restrict__ post_w,
                                            const float* __restrict__ post_b,
                                            float* __restrict__ ws) {
    __shared__ float red[256];
    int tid = threadIdx.x;
    int idx = blockIdx.x * 256 + tid;        // 262144 spatial positions
    int n   = idx >> 12;
    int pos = idx & 4095;
    const float* y2 = ws + WS_Y2;
    float h[4];
#pragma unroll
    for (int ic = 0; ic < 4; ++ic) {
        float raw = y2[(n * 4 + ic) * 4096 + pos];
        h[ic] = fmaxf(raw * ws[WS_SC2 + 2 * ic] + ws[WS_SC2 + 2 * ic + 1], 0.0f);
    }
    float z0 = pre_b[0], z1 = pre_b[1];
#pragma unroll
    for (int ic = 0; ic < 4; ++ic) { z0 += pre_w[ic] * h[ic]; z1 += pre_w[4 + ic] * h[ic]; }
    float best = 1e30f; int bi = 0;
#pragma unroll
    for (int k = 0; k < 3; ++k) {            // strict < => first minimum, matches jnp.argmin
        float d0 = z0 - cb[2 * k], d1 = z1 - cb[2 * k + 1];
        float d = d0 * d0 + d1 * d1;
        if (d < best) { best = d; bi = k; }
    }
    float q0 = cb[2 * bi], q1 = cb[2 * bi + 1];
    float loc = (q0 - z0) * (q0 - z0) + (q1 - z1) * (q1 - z1);
    float s0 = z0 + (q0 - z0);               // straight-through forward value
    float s1 = z1 + (q1 - z1);
    float* h3 = ws + WS_H3;
#pragma unroll
    for (int oc = 0; oc < 4; ++oc)
        h3[(n * 4 + oc) * 4096 + pos] = post_b[oc] + post_w[2 * oc] * s0 + post_w[2 * oc + 1] * s1;
    red[tid] = loc;
    __syncthreads();
    for (int off = 128; off > 0; off >>= 1) { if (tid < off) red[tid] += red[tid + off]; __syncthreads(); }
    if (tid == 0) ws[WS_LOSSP + blockIdx.x] = red[0];
}

__global__ void k_lossfin(const float* __restrict__ ws, float* __restrict__ loss_out) {
    if (threadIdx.x == 0) {
        float s = 0.f;
        for (int i = 0; i < 1024; ++i) s += ws[WS_LOSSP + i];
        // (codebook + BETA*commitment) = 1.2 * mean over 64*4096*2 elements
        *loss_out = s * (1.2f / 524288.0f);
    }
}

// ===== dec1: ConvTranspose 4->16, k4 s2 p1 (as dilated conv pad 2), phase-decomposed =====
// Per wave: fixed (n, oy, px), M=16 same-parity ox, N=16 oc, K=16 (4ic x 2x2 taps)
__global__ __launch_bounds__(256) void k_dec1(const float* __restrict__ w,
                                              const float* __restrict__ bias,
                                              float* __restrict__ ws) {
    __shared__ float lw[1024];               // dec_w1[oc][ic][ky][kx]
    int tid = threadIdx.x;
    for (int i = tid; i < 1024; i += 256) lw[i] = w[i];
    __syncthreads();
    int wave = blockIdx.x * 8 + (tid >> 5);  // 65536 waves
    int lane = tid & 31;
    int hi = lane >> 4;
    int l  = lane & 15;
    int jt = wave & 3;
    int px = (wave >> 2) & 1;
    int oy = (wave >> 3) & 127;
    int n  = wave >> 10;
    int py   = oy & 1;
    int irow = oy >> 1;
    const float* h3 = ws + WS_H3 + n * (4 * 64 * 64);
    v8f acc = {};
#pragma unroll
    for (int s = 0; s < 4; ++s) {
        int kb = 4 * s + 2 * hi;
        v2f a, bf;
#pragma unroll
        for (int u = 0; u < 2; ++u) {
            int k  = kb + u;                 // k = ic*4 + ty*2 + tx
            int ic = k >> 2, ty = (k >> 1) & 1, tx = k & 1;
            int iy = irow + py - 1 + ty;
            int j  = jt * 16 + l;
            int ix = j + px - 1 + tx;
            bool inb = ((unsigned)iy < 64u) & ((unsigned)ix < 64u);
            float v = h3[(ic * 64 + clampi(iy, 0, 63)) * 64 + clampi(ix, 0, 63)];
            a[u] = inb ? v : 0.0f;
            int ky = py + 2 * ty, kx = px + 2 * tx;
            bf[u] = lw[l * 64 + ic * 16 + ky * 4 + kx];
        }
        acc = __builtin_amdgcn_wmma_f32_16x16x4_f32(false, a, false, bf, (short)0, acc, false, false);
    }
    float bi = bias[l];
    float* y3 = ws + WS_BUF1;
    int rowbase = ((n * 16 + l) * 128 + oy) * 128 + px + 2 * (jt * 16);
#pragma unroll
    for (int r = 0; r < 8; ++r) y3[rowbase + 2 * (r + 8 * hi)] = acc[r] + bi;
}

// ===== dec2: ConvTranspose 16->1, BN3+ReLU fused; Cout=1 so VALU (WMMA would be 1/16 used) =====
__global__ __launch_bounds__(256) void k_dec2(const float* __restrict__ w,
                                              const float* __restrict__ bias,
                                              const float* __restrict__ ws,
                                              float* __restrict__ out) {
    __shared__ float lw[256], lsc[16], lsh[16];
    int tid = threadIdx.x;
    lw[tid] = w[tid];                        // dec_w2[ic][ky][kx]
    if (tid < 16) { lsc[tid] = ws[WS_SC3 + 2 * tid]; lsh[tid] = ws[WS_SC3 + 2 * tid + 1]; }
    __syncthreads();
    int idx = blockIdx.x * 256 + tid;        // 64*256*256 outputs
    int n   = idx >> 16;
    int rem = idx & 65535;
    int oy  = rem >> 8;
    int ox  = rem & 255;
    int py = oy & 1, px = ox & 1;
    int i = oy >> 1, j = ox >> 1;
    const float* y3 = ws + WS_BUF1 + n * (16 * 128 * 128);
    float acc = bias[0];
#pragma unroll 4
    for (int ic = 0; ic < 16; ++ic) {
        float sc = lsc[ic], sh = lsh[ic];
#pragma unroll
        for (int ty = 0; ty < 2; ++ty) {
            int iy = i + py - 1 + ty;
            bool iny = (unsigned)iy < 128u;
#pragma unroll
            for (int tx = 0; tx < 2; ++tx) {
                int ix = j + px - 1 + tx;
                bool inb = iny & ((unsigned)ix < 128u);
                float raw = y3[(ic * 128 + clampi(iy, 0, 127)) * 128 + clampi(ix, 0, 127)];
                float v = fmaxf(raw * sc + sh, 0.f);
                acc += (inb ? v : 0.0f) * lw[ic * 16 + (py + 2 * ty) * 4 + (px + 2 * tx)];
            }
        }
    }
    out[idx] = acc;
}

// ===================================== launch =====================================
extern "C" void kernel_launch(void* const* d_in, const int* in_sizes, int n_in,
                              void* d_out, int out_size, void* d_ws, size_t ws_size,
                              hipStream_t stream) {
    const float* x      = (const float*)d_in[0];
    const float* enc_w1 = (const float*)d_in[1];
    const float* enc_b1 = (const float*)d_in[2];
    const float* bn1_g  = (const float*)d_in[3];
    const float* bn1_b  = (const float*)d_in[4];
    const float* enc_w2 = (const float*)d_in[5];
    const float* enc_b2 = (const float*)d_in[6];
    const float* bn2_g  = (const float*)d_in[7];
    const float* bn2_b  = (const float*)d_in[8];
    const float* pre_w  = (const float*)d_in[9];
    const float* pre_b  = (const float*)d_in[10];
    const float* cbk    = (const float*)d_in[11];
    const float* post_w = (const float*)d_in[12];
    const float* post_b = (const float*)d_in[13];
    const float* dec_w1 = (const float*)d_in[14];
    const float* dec_b1 = (const float*)d_in[15];
    const float* bn3_g  = (const float*)d_in[16];
    const float* bn3_b  = (const float*)d_in[17];
    const float* dec_w2 = (const float*)d_in[18];
    const float* dec_b2 = (const float*)d_in[19];
    float* wsf  = (float*)d_ws;
    float* outf = (float*)d_out;

    // 1) conv1 (WMMA) -> y1
    k_conv1<<<8192, 256, 0, stream>>>(x, enc_w1, enc_b1, wsf);
    // 2) BN1 stats (16 ch, PS=128*128, P=128 blocks/ch)
    k_stats<<<16 * 128, 256, 0, stream>>>(wsf + WS_BUF1, wsf + WS_STATS, 16, 14, 128);
    k_bnfin<<<1, 32, 0, stream>>>(wsf + WS_STATS, bn1_g, bn1_b, wsf + WS_SC1, 16, 128, 1.0f / 1048576.0f);
    // 3) conv2 (WMMA, BN1+ReLU fused) -> y2
    k_conv2<<<2048, 256, 0, stream>>>(enc_w2, enc_b2, wsf);
    // 4) BN2 stats (4 ch, PS=64*64, P=64 blocks/ch)
    k_stats<<<4 * 64, 256, 0, stream>>>(wsf + WS_Y2, wsf + WS_STATS, 4, 12, 64);
    k_bnfin<<<1, 32, 0, stream>>>(wsf + WS_STATS, bn2_g, bn2_b, wsf + WS_SC2, 4, 64, 1.0f / 262144.0f);
    // 5) fused bottleneck + VQ -> h3, loss partials
    k_vq<<<1024, 256, 0, stream>>>(pre_w, pre_b, cbk, post_w, post_b, wsf);
    k_lossfin<<<1, 32, 0, stream>>>(wsf, outf + 4194304);
    // 6) dec1 (WMMA) -> y3 (reuses y1 buffer)
    k_dec1<<<8192, 256, 0, stream>>>(dec_w1, dec_b1, wsf);
    // 7) BN3 stats
    k_stats<<<16 * 128, 256, 0, stream>>>(wsf + WS_BUF1, wsf + WS_STATS, 16, 14, 128);
    k_bnfin<<<1, 32, 0, stream>>>(wsf + WS_STATS, bn3_g, bn3_b, wsf + WS_SC3, 16, 128, 1.0f / 1048576.0f);
    // 8) dec2 (BN3+ReLU fused) -> image output
    k_dec2<<<16384, 256, 0, stream>>>(dec_w2, dec_b2, wsf, outf);
}